// DigitConvolutionalModel_51367808860624
// MI455X (gfx1250) — compile-verified
//
#include <hip/hip_runtime.h>

typedef float v2f __attribute__((ext_vector_type(2)));
typedef float v8f __attribute__((ext_vector_type(8)));
typedef unsigned int u32x4 __attribute__((ext_vector_type(4)));
typedef int i32x8 __attribute__((ext_vector_type(8)));
typedef int i32x4 __attribute__((ext_vector_type(4)));

#define TB      16      // batch rows per workgroup
#define IMG_N   784     // 28*28
#define HK      676     // 26*26 conv features (== 169 * 4)
#define N1      128
#define HSTR    676     // h LDS row stride in dwords (4*169, 169 coprime 16 -> bank-free)
#define H2STR   132     // h2 LDS row stride in dwords (4*33)
#define NTHREADS 256

#if defined(__HIP_DEVICE_COMPILE__) &&                      \
    __has_builtin(__builtin_amdgcn_tensor_load_to_lds) &&   \
    __has_builtin(__builtin_amdgcn_s_wait_tensorcnt)
#define USE_TDM 1
#else
#define USE_TDM 0
#endif

#define WMMA_F32(a, b, c) \
    __builtin_amdgcn_wmma_f32_16x16x4_f32(false, (a), false, (b), (short)0, (c), false, false)

__global__ __launch_bounds__(NTHREADS)
void digit_mlp_fused(const float* __restrict__ x,
                     const float* __restrict__ cw,
                     const float* __restrict__ w1,
                     const float* __restrict__ b1,
                     const float* __restrict__ w2,
                     const float* __restrict__ b2,
                     float* __restrict__ out)
{
    extern __shared__ float lds[];
    float* simg = lds;                    // 16 * 784
    float* sh   = simg + TB * IMG_N;      // 16 * 676 (stride HSTR)
    float* sh2  = sh   + TB * HSTR;       // 16 * 132 (stride H2STR)

    const int tid  = threadIdx.x;
    const int lane = tid & 31;
    const int wave = tid >> 5;
    const int row0 = blockIdx.x * TB;

    // ---- Stage 1: stage 16x784 image tile into LDS ----
#if USE_TDM
    {
        // Each of the 8 waves DMAs its 2 rows (1568 contiguous fp32) via the TDM.
        const unsigned w = (unsigned)__builtin_amdgcn_readfirstlane(tid >> 5);
        const unsigned ROWF = 2u * IMG_N; // 1568 elements per wave
        const unsigned long long ga =
            (unsigned long long)(const void*)(x + (size_t)row0 * IMG_N + (size_t)w * ROWF);

        u32x4 g0;
        g0.x = 0x1u;                                 // count=1, user mode
        g0.y = w * (ROWF * 4u);                      // lds_addr (dyn LDS base == 0)
        g0.z = (unsigned)(ga & 0xFFFFFFFFull);       // global_addr[31:0]
        g0.w = (unsigned)((ga >> 32) & 0x1FFFFFFull) // global_addr[56:32]
             | (2u << 30);                           // type = 2 ("image")

        i32x8 g1;
        g1[0] = (2 << 16);                           // data_size = 4 bytes
        g1[1] = (int)(ROWF << 16);                   // tensor_dim0 = 1568 (bits 79:48)
        g1[2] = (1 << 16);                           // tensor_dim1 = 1
        g1[3] = (int)(ROWF << 16);                   // tile_dim0 = 1568 (bits 127:112)
        g1[4] = 1;                                   // tile_dim1 = 1
        g1[5] = (int)ROWF;                           // tensor_dim0_stride = 1568
        g1[6] = 0;
        g1[7] = 0;

        i32x4 gz4 = {0, 0, 0, 0};                    // groups 2/3 unused (<=2D)
        i32x8 gz8 = {0, 0, 0, 0, 0, 0, 0, 0};        // trailing group (unused)
        __builtin_amdgcn_tensor_load_to_lds(g0, g1, gz4, gz4, gz8, 0);
        __builtin_amdgcn_s_wait_tensorcnt((short)0);
    }
#else
    {
        const float4* xin = (const float4*)(x + (size_t)row0 * IMG_N);
        float4* dst = (float4*)simg;
        #pragma unroll 4
        for (int i = tid; i < TB * (IMG_N / 4); i += NTHREADS)
            dst[i] = xin[i];
    }
#endif
    // conv weights (tiny, L2 hot)
    const float c0 = cw[0], c1 = cw[1], c2 = cw[2];
    const float c3 = cw[3], c4 = cw[4], c5 = cw[5];
    const float c6 = cw[6], c7 = cw[7], c8 = cw[8];
    __syncthreads();

    // ---- Stage 2: 3x3 valid cross-correlation, 28x28 -> 26x26, into sh ----
    for (int i = tid; i < TB * HK; i += NTHREADS) {
        int s = i / HK;
        int o = i - s * HK;
        int r = o / 26;
        int c = o - r * 26;
        const float* ip = simg + s * IMG_N + r * 28 + c;
        float acc = ip[0]  * c0 + ip[1]  * c1 + ip[2]  * c2
                  + ip[28] * c3 + ip[29] * c4 + ip[30] * c5
                  + ip[56] * c6 + ip[57] * c7 + ip[58] * c8;
        sh[s * HSTR + o] = acc;
    }
    __syncthreads();

    // ---- Stage 3: FC1 via V_WMMA_F32_16X16X4_F32, K = 676, dual acc chains ----
    // A (16x4 f32): lane m=lane&15 holds h[m][k0+kp..+1], kp=2*(lane>>4)
    // B (4x16 f32): lane n=lane&15 holds w1[k0+kp..+1][col]
    // C/D (16x16):  VGPR i, lane l -> (M = i + 8*(l>>4), N = l&15)
    {
        const int m   = lane & 15;
        const int kp  = (lane >> 4) * 2;
        const int col = wave * 16 + m;
        const float* hrow = sh + m * HSTR + kp;
        const float* wcol = w1 + (size_t)kp * N1 + col;

        v8f acc0 = {};
        v8f acc1 = {};
        int k0 = 0;
        for (; k0 + 8 <= HK; k0 += 8) {
            v2f a0, b0;
            a0.x = hrow[k0];
            a0.y = hrow[k0 + 1];
            b0.x = wcol[(size_t)k0 * N1];
            b0.y = wcol[(size_t)(k0 + 1) * N1];

            v2f a1, b1v;
            a1.x = hrow[k0 + 4];
            a1.y = hrow[k0 + 5];
            b1v.x = wcol[(size_t)(k0 + 4) * N1];
            b1v.y = wcol[(size_t)(k0 + 5) * N1];

            acc0 = WMMA_F32(a0, b0, acc0);
            acc1 = WMMA_F32(a1, b1v, acc1);
        }
        if (k0 < HK) {              // K tail: 676 = 84*8 + 4
            v2f a, b;
            a.x = hrow[k0];
            a.y = hrow[k0 + 1];
            b.x = wcol[(size_t)k0 * N1];
            b.y = wcol[(size_t)(k0 + 1) * N1];
            acc0 = WMMA_F32(a, b, acc0);
        }
        v8f acc = acc0 + acc1;

        // bias + ReLU from C layout, scatter into sh2[16][128] (stride 132)
        const int n     = lane & 15;
        const int mbase = (lane >> 4) * 8;
        const float bn  = b1[wave * 16 + n];
        #pragma unroll
        for (int i = 0; i < 8; ++i) {
            float v = acc[i] + bn;
            v = v > 0.0f ? v : 0.0f;
            sh2[(mbase + i) * H2STR + wave * 16 + n] = v;
        }
    }
    __syncthreads();

    // ---- Stage 4: FC2 via WMMA, K = 128, N padded 10 -> 16 with zeros ----
    // All waves compute redundantly (keeps EXEC all-ones for WMMA); wave 0 stores.
    {
        const int m  = lane & 15;
        const int kp = (lane >> 4) * 2;
        const int n  = lane & 15;
        const bool nvalid = (n < 10);

        v8f acc0 = {};
        v8f acc1 = {};
        for (int k0 = 0; k0 < N1; k0 += 8) {
            const float* h2row = sh2 + m * H2STR + k0 + kp;
            v2f a0, b0;
            a0.x = h2row[0];
            a0.y = h2row[1];
            b0.x = nvalid ? w2[(k0 + kp)     * 10 + n] : 0.0f;
            b0.y = nvalid ? w2[(k0 + kp + 1) * 10 + n] : 0.0f;

            v2f a1, b1v;
            a1.x = h2row[4];
            a1.y = h2row[5];
            b1v.x = nvalid ? w2[(k0 + kp + 4) * 10 + n] : 0.0f;
            b1v.y = nvalid ? w2[(k0 + kp + 5) * 10 + n] : 0.0f;

            acc0 = WMMA_F32(a0, b0, acc0);
            acc1 = WMMA_F32(a1, b1v, acc1);
        }
        v8f acc = acc0 + acc1;

        if (wave == 0 && nvalid) {
            const float bn  = b2[n];
            const int mbase = (lane >> 4) * 8;
            #pragma unroll
            for (int i = 0; i < 8; ++i)
                out[(size_t)(row0 + mbase + i) * 10 + n] = acc[i] + bn;
        }
    }
}

extern "C" void kernel_launch(void* const* d_in, const int* in_sizes, int n_in,
                              void* d_out, int out_size, void* d_ws, size_t ws_size,
                              hipStream_t stream)
{
    const float* x   = (const float*)d_in[0];
    const float* cw  = (const float*)d_in[1];
    const float* w1  = (const float*)d_in[2];
    const float* b1  = (const float*)d_in[3];
    const float* w2  = (const float*)d_in[4];
    const float* b2  = (const float*)d_in[5];
    float* out = (float*)d_out;

    const int B = in_sizes[0] / IMG_N;          // 16384
    const int nblocks = (B + TB - 1) / TB;      // 1024
    const size_t shmem =
        (size_t)(TB * IMG_N + TB * HSTR + TB * H2STR) * sizeof(float); // ~99.5 KB

    digit_mlp_fused<<<nblocks, NTHREADS, shmem, stream>>>(x, cw, w1, b1, w2, b2, out);
}